// AttentionBlock_54468775248479
// MI455X (gfx1250) — compile-verified
//
#include <hip/hip_runtime.h>

typedef __attribute__((ext_vector_type(2))) float v2f;
typedef __attribute__((ext_vector_type(8))) float v8f;

#define B_   4
#define C_   256
#define N_   4096
#define CIN  258

// D = A(16x4 f32) * B(4x16 f32) + C(16x16 f32), one matrix per wave32.
static __device__ __forceinline__ v8f wmma_f32(v2f a, v2f b, v8f c) {
  return __builtin_amdgcn_wmma_f32_16x16x4_f32(false, a, false, b, (short)0, c,
                                               false, false);
}

static __device__ __forceinline__ float rmax16(float v) {
  v = fmaxf(v, __shfl_xor(v, 1, 32));
  v = fmaxf(v, __shfl_xor(v, 2, 32));
  v = fmaxf(v, __shfl_xor(v, 4, 32));
  v = fmaxf(v, __shfl_xor(v, 8, 32));
  return v;
}
static __device__ __forceinline__ float rsum16(float v) {
  v += __shfl_xor(v, 1, 32);
  v += __shfl_xor(v, 2, 32);
  v += __shfl_xor(v, 4, 32);
  v += __shfl_xor(v, 8, 32);
  return v;
}

// ---------------- Kernel A: q/k/v = relu(W @ [x;pos] + b) ----------------
// grid (N/16, C/16, B*3), block 32. One wave -> one 16(o) x 16(n) tile.
__global__ __launch_bounds__(32) void qkv_kernel(
    const float* __restrict__ x,
    const float* __restrict__ qw, const float* __restrict__ qb,
    const float* __restrict__ kw, const float* __restrict__ kb,
    const float* __restrict__ vw, const float* __restrict__ vb,
    float* __restrict__ qo, float* __restrict__ ko, float* __restrict__ vo) {
  const int lane = threadIdx.x;
  const int r  = lane & 15;   // A-row / B-col / D-col index
  const int hf = lane >> 4;   // K sub-pair select
  const int n0 = blockIdx.x * 16;
  const int o0 = blockIdx.y * 16;
  const int b   = blockIdx.z / 3;
  const int mat = blockIdx.z % 3;
  const float* w; const float* bias; float* out;
  if (mat == 0)      { w = qw; bias = qb; out = qo; }
  else if (mat == 1) { w = kw; bias = kb; out = ko; }
  else               { w = vw; bias = vb; out = vo; }

  const float* xb = x + (size_t)b * C_ * N_;
  const int n = n0 + r;
  v8f acc = {};
#pragma unroll 8
  for (int kk = 0; kk < 64; ++kk) {
    const int c = kk * 4 + hf * 2;
    v2f a = *(const v2f*)(w + (size_t)(o0 + r) * CIN + c);   // A[o][c], c even -> 8B aligned
    v2f bb;
    bb.x = xb[(size_t)c * N_ + n];                           // B[c][n]
    bb.y = xb[(size_t)(c + 1) * N_ + n];
    acc = wmma_f32(a, bb, acc);
  }
  // positional channels 256/257 + bias + relu
  const float px = (float)(n >> 6) * (1.0f / 64.0f);
  const float py = (float)(n & 63) * (1.0f / 64.0f);
  float* ob = out + (size_t)b * C_ * N_;
#pragma unroll
  for (int j = 0; j < 8; ++j) {
    const int o = o0 + j + hf * 8;                           // D row
    v2f wp = *(const v2f*)(w + (size_t)o * CIN + 256);
    float val = acc[j] + wp.x * px + wp.y * py + bias[o];
    ob[(size_t)o * N_ + n] = fmaxf(val, 0.0f);
  }
}

// ---------------- Kernel B: causal flash attention ----------------
// grid (N/16, B), block 32. One wave -> 16 query rows, full V=256 cols.
__global__ __launch_bounds__(32) void attn_kernel(
    const float* __restrict__ qbuf, const float* __restrict__ kbuf,
    const float* __restrict__ vbuf, float* __restrict__ abuf) {
  __shared__ float qs[16 * CIN];   // q tile, A-operand friendly, stride 258
  __shared__ float pt[16 * 18];    // P transpose staging, stride 18
  const int lane = threadIdx.x;
  const int r  = lane & 15;
  const int hf = lane >> 4;
  const int nt = blockIdx.x;
  const int b  = blockIdx.y;
  const int n0 = nt * 16;

  const float* qp = qbuf + (size_t)b * C_ * N_;
  const float* kp = kbuf + (size_t)b * C_ * N_;
  const float* vp = vbuf + (size_t)b * C_ * N_;

  for (int i = lane; i < 16 * 256; i += 32) {
    const int rr = i & 15;
    const int cc = i >> 4;
    qs[rr * CIN + cc] = qp[(size_t)cc * N_ + (n0 + rr)];
  }
  __syncthreads();

  v8f zerov = {};
  v8f acc[16];
#pragma unroll
  for (int t = 0; t < 16; ++t) acc[t] = zerov;
  float m_run[8], l_run[8];
#pragma unroll
  for (int j = 0; j < 8; ++j) { m_run[j] = -3.0e38f; l_run[j] = 0.0f; }

  for (int mt = 0; mt <= nt; ++mt) {
    const int m0 = mt * 16;
    const int mcol = m0 + r;
    // scores S = q^T k  (16 n-rows x 16 m-cols), K = 256
    v8f s = {};
#pragma unroll 8
    for (int kk = 0; kk < 64; ++kk) {
      const int c = kk * 4 + hf * 2;
      v2f a = *(const v2f*)(&qs[r * CIN + c]);               // A[n][c] = q[c][n]
      v2f bb;
      bb.x = kp[(size_t)c * N_ + mcol];                      // B[c][m]
      bb.y = kp[(size_t)(c + 1) * N_ + mcol];
      s = wmma_f32(a, bb, s);
    }
    if (mt == nt) {                                          // causal mask on diagonal tile
#pragma unroll
      for (int j = 0; j < 8; ++j) {
        const int rowr = j + hf * 8;
        s[j] = (r > rowr) ? -3.0e38f : s[j];
      }
    }
    // online softmax over this 16-col slab
    float scale[8];
#pragma unroll
    for (int j = 0; j < 8; ++j) {
      const float rm = rmax16(s[j]);
      const float mn = fmaxf(m_run[j], rm);
      scale[j] = __expf(m_run[j] - mn);
      m_run[j] = mn;
      const float p = __expf(s[j] - mn);
      s[j] = p;
      l_run[j] = l_run[j] * scale[j] + rsum16(p);
    }
#pragma unroll
    for (int t = 0; t < 16; ++t)
#pragma unroll
      for (int j = 0; j < 8; ++j) acc[t][j] *= scale[j];

    // transpose P (D layout -> A layout) through LDS
    __syncthreads();
#pragma unroll
    for (int j = 0; j < 8; ++j) pt[(j + hf * 8) * 18 + r] = s[j];
    __syncthreads();
    v2f ap[4];
#pragma unroll
    for (int k2 = 0; k2 < 4; ++k2)
      ap[k2] = *(const v2f*)(&pt[r * 18 + k2 * 4 + hf * 2]);

    // acc += P @ v^T   (A = P 16x16 over m, B = v^T[m][v])
#pragma unroll
    for (int t = 0; t < 16; ++t) {
      const int vr = t * 16 + r;                             // v column index
#pragma unroll
      for (int k2 = 0; k2 < 4; ++k2) {
        const int mm = m0 + k2 * 4 + hf * 2;
        v2f bb = *(const v2f*)(&vp[(size_t)vr * N_ + mm]);   // contiguous in m
        acc[t] = wmma_f32(ap[k2], bb, acc[t]);
      }
    }
  }

  float inv[8];
#pragma unroll
  for (int j = 0; j < 8; ++j) inv[j] = 1.0f / l_run[j];
  float* ab = abuf + (size_t)b * N_ * C_;                    // [n][v], coalesced stores
#pragma unroll
  for (int t = 0; t < 16; ++t) {
    const int v = t * 16 + r;
#pragma unroll
    for (int j = 0; j < 8; ++j) {
      const int n = n0 + j + hf * 8;
      ab[(size_t)n * C_ + v] = acc[t][j] * inv[j];
    }
  }
}

// ---------------- Kernel C: out = x + relu(fc_w @ attn_out + fc_b) ----------------
// grid (N/16, C/16, B), block 32.
__global__ __launch_bounds__(32) void fc_kernel(
    const float* __restrict__ x, const float* __restrict__ fcw,
    const float* __restrict__ fcb, const float* __restrict__ abuf,
    float* __restrict__ out) {
  const int lane = threadIdx.x;
  const int r  = lane & 15;
  const int hf = lane >> 4;
  const int n0 = blockIdx.x * 16;
  const int o0 = blockIdx.y * 16;
  const int b  = blockIdx.z;
  const float* ab = abuf + (size_t)b * N_ * C_;
  const int n = n0 + r;
  v8f acc = {};
#pragma unroll 8
  for (int kk = 0; kk < 64; ++kk) {
    const int c = kk * 4 + hf * 2;
    v2f a  = *(const v2f*)(fcw + (size_t)(o0 + r) * C_ + c); // A[o][v]
    v2f bb = *(const v2f*)(&ab[(size_t)n * C_ + c]);         // B[v][n] from [n][v] layout
    acc = wmma_f32(a, bb, acc);
  }
  const float* xb = x + (size_t)b * C_ * N_;
  float* ob = out + (size_t)b * C_ * N_;
#pragma unroll
  for (int j = 0; j < 8; ++j) {
    const int o = o0 + j + hf * 8;
    const float val = fmaxf(acc[j] + fcb[o], 0.0f) + xb[(size_t)o * N_ + n];
    ob[(size_t)o * N_ + n] = val;
  }
}

extern "C" void kernel_launch(void* const* d_in, const int* in_sizes, int n_in,
                              void* d_out, int out_size, void* d_ws, size_t ws_size,
                              hipStream_t stream) {
  (void)in_sizes; (void)n_in; (void)out_size; (void)ws_size;
  const float* x   = (const float*)d_in[0];
  const float* qw  = (const float*)d_in[1];
  const float* qb  = (const float*)d_in[2];
  const float* kw  = (const float*)d_in[3];
  const float* kb  = (const float*)d_in[4];
  const float* vw  = (const float*)d_in[5];
  const float* vb  = (const float*)d_in[6];
  const float* fcw = (const float*)d_in[7];
  const float* fcb = (const float*)d_in[8];
  float* out = (float*)d_out;
  float* ws  = (float*)d_ws;
  const size_t mat = (size_t)B_ * C_ * N_;  // 4M floats = 16 MB
  float* qbuf = ws;
  float* kbuf = ws + mat;
  float* vbuf = ws + 2 * mat;
  float* abuf = ws + 3 * mat;

  qkv_kernel<<<dim3(N_ / 16, C_ / 16, B_ * 3), dim3(32), 0, stream>>>(
      x, qw, qb, kw, kb, vw, vb, qbuf, kbuf, vbuf);
  attn_kernel<<<dim3(N_ / 16, B_), dim3(32), 0, stream>>>(qbuf, kbuf, vbuf, abuf);
  fc_kernel<<<dim3(N_ / 16, C_ / 16, B_), dim3(32), 0, stream>>>(
      x, fcw, fcb, abuf, out);
}